// AttentionHead_84713934946555
// MI455X (gfx1250) — compile-verified
//
#include <hip/hip_runtime.h>

// ---------------------------------------------------------------------------
// Fused causal attention head with RoPE for MI455X (gfx1250, wave32).
//   B=4, T=4096, C=1024, H=64, fp32.
// Matrix math: V_WMMA_F32_16X16X4_F32 (fp32-accurate tensor path).
// LDS staging: GLOBAL_LOAD_ASYNC_TO_LDS_B128 (ASYNCcnt), double-buffered.
// Weights pre-transposed (N x K) so every B-fragment lane-pair is contiguous.
// ---------------------------------------------------------------------------

typedef __attribute__((ext_vector_type(2))) float v2f;
typedef __attribute__((ext_vector_type(8))) float v8f;

constexpr int B = 4;
constexpr int T = 4096;
constexpr int C = 1024;
constexpr int H = 64;
constexpr int ROWS = B * T;          // 16384 rows of x / q / k / v
constexpr int WPB  = 4;              // waves per block in the attention kernel

// Issue one 16-byte async global->LDS copy (no VGPR round trip, ASYNCcnt).
__device__ __forceinline__ void async_copy_b128(void* lds_dst, const void* gsrc) {
  const unsigned ldsOff = (unsigned)(size_t)lds_dst;  // flat addr low 32 = LDS offset
  asm volatile("global_load_async_to_lds_b128 %0, %1, off"
               :: "v"(ldsOff), "v"(gsrc) : "memory");
}

__device__ __forceinline__ void wait_async0() {
  asm volatile("s_wait_asynccnt 0" ::: "memory");
}

// A-fragment loader for a row-major M x K tile (also serves as the B-fragment
// loader when B is stored transposed, i.e. as N x K row-major).
//   lane 0..15  -> row = lane,    cols k0, k0+1
//   lane 16..31 -> row = lane-16, cols k0+2, k0+3
__device__ __forceinline__ v2f frag_nk(const float* __restrict__ base, int stride, int k0) {
  const int lane = threadIdx.x & 31;
  const int r  = lane & 15;
  const int kk = k0 + ((lane >> 4) << 1);
  v2f f;
  f.x = base[r * stride + kk];
  f.y = base[r * stride + kk + 1];
  return f;
}

// B-fragment loader when B is stored K x N row-major (stride = row pitch).
__device__ __forceinline__ v2f frag_kn(const float* __restrict__ base, int stride, int k0, int nb) {
  const int lane = threadIdx.x & 31;
  const int n  = lane & 15;
  const int kk = k0 + ((lane >> 4) << 1);
  v2f f;
  f.x = base[kk * stride + nb + n];
  f.y = base[(kk + 1) * stride + nb + n];
  return f;
}

// ---------------------------------------------------------------------------
// Kernel 0: transpose Wq/Wk/Wv (C x H) into Wt (3 x H x C).  768 KB once;
// makes every projection B-fragment a contiguous 8-byte load per lane.
// ---------------------------------------------------------------------------
__global__ void wt_kernel(const float* __restrict__ Wq, const float* __restrict__ Wk,
                          const float* __restrict__ Wv, float* __restrict__ Wt) {
  const int idx = blockIdx.x * blockDim.x + threadIdx.x;
  if (idx >= 3 * C * H) return;
  const int m = idx / (C * H);
  const int r = idx % (C * H);
  const int k = r / H;                              // consecutive idx -> coalesced read
  const int n = r % H;
  const float* __restrict__ W = (m == 0) ? Wq : (m == 1 ? Wk : Wv);
  Wt[(size_t)m * C * H + (size_t)n * C + k] = W[(size_t)k * H + n];
}

// ---------------------------------------------------------------------------
// Kernel 1: q/k/v projections.  One block per 16-row tile of x.
// 12 waves: wave -> (matrix q/k/v, 16-col tile of H).
// x staged in 16 x 256 chunks, double-buffered via async-to-LDS.
// ---------------------------------------------------------------------------
constexpr int KC = 256;              // K-chunk (16*256*4B = 16 KB per buffer)

__global__ __launch_bounds__(384) void qkv_kernel(
    const float* __restrict__ x, const float* __restrict__ Wt,
    float* __restrict__ q, float* __restrict__ k, float* __restrict__ v) {
  __shared__ __align__(16) float lx[2][16 * KC];

  const int tid  = threadIdx.x;
  const int row0 = blockIdx.x * 16;

  const int wave = tid >> 5;         // 0..11
  const int mat  = wave >> 2;        // 0:q 1:k 2:v
  const int ct   = wave & 3;         // 16-col tile of H
  // N x K weight tile for this wave: rows ct*16 .. ct*16+15, K = C
  const float* __restrict__ Wm = Wt + (size_t)mat * C * H + (size_t)(ct * 16) * C;
  float* __restrict__       out = (mat == 0) ? q : (mat == 1 ? k : v);

  // stage the 16 x KC chunk `cidx` into buffer `buf` (1024 float4 transfers)
  auto stage = [&](int cidx, int buf) {
    const int kc = cidx * KC;
    for (int i = tid; i < 16 * KC / 4; i += 384) {
      const int r  = i / (KC / 4);
      const int c4 = i % (KC / 4);
      async_copy_b128(&lx[buf][i * 4],
                      x + (size_t)(row0 + r) * C + kc + c4 * 4);
    }
  };

  stage(0, 0);

  v8f acc = {};
  for (int cidx = 0; cidx < C / KC; ++cidx) {
    wait_async0();                   // my async writes to lx[cidx&1] done
    __syncthreads();                 // everyone's done -> buffer valid
    if (cidx + 1 < C / KC) stage(cidx + 1, (cidx + 1) & 1);  // overlap next chunk

    const float* __restrict__ lxb = lx[cidx & 1];
#pragma unroll 8
    for (int k0 = 0; k0 < KC; k0 += 4) {
      v2f a = frag_nk(lxb, KC, k0);                // A: 16x4 slice of x-tile
      v2f b = frag_nk(Wm, C, cidx * KC + k0);      // B: 4x16 slice of Wt (NxK)
      acc = __builtin_amdgcn_wmma_f32_16x16x4_f32(false, a, false, b,
                                                  (short)0, acc, false, false);
    }
  }

  // D layout: vgpr r -> M = r + 8*(lane>=16), N = lane&15
  const int lane = tid & 31;
  const int n  = lane & 15;
  const int mb = (lane >> 4) << 3;
#pragma unroll
  for (int r = 0; r < 8; ++r)
    out[(size_t)(row0 + mb + r) * H + ct * 16 + n] = acc[r];
}

// ---------------------------------------------------------------------------
// Kernel 2: RoPE on q and k (in workspace).  One thread per rotated pair.
// ---------------------------------------------------------------------------
__global__ void rope_kernel(float* __restrict__ q, float* __restrict__ k) {
  const int NP = ROWS * (H / 2);                   // pairs per tensor
  const int idx = blockIdx.x * blockDim.x + threadIdx.x;
  if (idx >= 2 * NP) return;
  float* base = (idx < NP) ? q : k;
  const int p   = (idx < NP) ? idx : idx - NP;
  const int row = p >> 5;                          // / (H/2)
  const int h   = p & 31;
  const int t   = row & (T - 1);                   // position within sequence
  const float freq = __powf(10000.0f, -(float)(2 * h) / (float)H);
  float s, c;
  __sincosf((float)t * freq, &s, &c);
  float2* pr = (float2*)(base + (size_t)row * H) + h;
  const float2 val = *pr;
  float2 o;
  o.x = val.x * c - val.y * s;
  o.y = val.x * s + val.y * c;
  *pr = o;
}

// ---------------------------------------------------------------------------
// Kernel 3: flash attention.  4 waves / block; each wave owns one 16-row
// query tile; the block streams causal 16x64 K/V tiles through LDS with
// async-to-LDS double buffering (one barrier per key tile).
// ---------------------------------------------------------------------------
__global__ __launch_bounds__(32 * WPB) void attn_kernel(
    const float* __restrict__ q, const float* __restrict__ kk,
    const float* __restrict__ vv, float* __restrict__ out) {
  __shared__ __align__(16) float lq[WPB][16 * H];   // per-wave Q tile
  __shared__ __align__(16) float lk[2][16 * H];     // K tile ping-pong (NxK view)
  __shared__ __align__(16) float lv[2][16 * H];     // V tile ping-pong (KxN view)
  __shared__ __align__(16) float lp[WPB][16 * 16];  // per-wave P scratch

  const int tid  = threadIdx.x;
  const int wave = tid >> 5;
  const int lane = tid & 31;
  const int tilesPerBlock = T / (16 * WPB);
  const int b   = blockIdx.x / tilesPerBlock;
  const int qt  = blockIdx.x % tilesPerBlock;
  const int iw  = qt * WPB + wave;                 // this wave's query-tile idx
  const int qrow0 = iw * 16;                       // query row base (within batch)

  const float* __restrict__ qb = q  + (size_t)b * T * H;
  const float* __restrict__ kb = kk + (size_t)b * T * H;
  const float* __restrict__ vb = vv + (size_t)b * T * H;

  // stage K/V tile j into ping-pong buffer `buf` (exactly 4 B128 per thread: 2 K + 2 V)
  auto stage = [&](int j, int buf) {
    const float* ks = kb + (size_t)j * 16 * H;
    const float* vs = vb + (size_t)j * 16 * H;
    for (int i = tid; i < (16 * H) / 4; i += 32 * WPB) {
      async_copy_b128(&lk[buf][i * 4], ks + i * 4);
      async_copy_b128(&lv[buf][i * 4], vs + i * 4);
    }
  };

  // each wave loads its own Q tile (plain loads; done once)
  for (int i = lane; i < 16 * H; i += 32) lq[wave][i] = qb[(size_t)qrow0 * H + i];
  asm volatile("s_wait_dscnt 0" ::: "memory");     // cross-lane LDS visibility

  stage(0, 0);

  v8f o0 = {}, o1 = {}, o2 = {}, o3 = {};
  float m[8], l[8];
#pragma unroll
  for (int r = 0; r < 8; ++r) { m[r] = -3.0e38f; l[r] = 0.0f; }

  const int   jmax  = qt * WPB + (WPB - 1);        // uniform per block
  const float scale = 0.03125f;                    // C^-0.5 = 1/32

  for (int j = 0; j <= jmax; ++j) {
    wait_async0();                                 // my writes to buf[j&1] done
    __syncthreads();                               // all waves -> tile valid; and
                                                   // all reads of buf[(j+1)&1] from
                                                   // iter j-1 already consumed
    if (j < jmax) stage(j + 1, (j + 1) & 1);       // overlap next tile with compute
    if (j > iw) continue;                          // causal early-out (wave-uniform)

    const float* __restrict__ lkj = lk[j & 1];
    const float* __restrict__ lvj = lv[j & 1];

    // ---- S = Q * K^T  (16 wmma, K = 64) ----
    v8f s = {};
#pragma unroll
    for (int k0 = 0; k0 < H; k0 += 4) {
      v2f a  = frag_nk(lq[wave], H, k0);
      v2f bf = frag_nk(lkj, H, k0);                // K tile is N x K row-major
      s = __builtin_amdgcn_wmma_f32_16x16x4_f32(false, a, false, bf,
                                                (short)0, s, false, false);
    }

    // ---- online softmax in D-layout (rows striped r / r+8, cols = lanes) ----
    const int n    = lane & 15;
    const int mb   = (lane >> 4) << 3;
    const int kcol = j * 16 + n;
    float alpha[8];
#pragma unroll
    for (int r = 0; r < 8; ++r) {
      float sv = s[r] * scale;
      const int qrow = qrow0 + mb + r;
      if (kcol > qrow) sv = -3.0e38f;              // causal mask
      float rm = sv;                               // row max over 16-lane half
      rm = fmaxf(rm, __shfl_xor(rm, 1, 32));
      rm = fmaxf(rm, __shfl_xor(rm, 2, 32));
      rm = fmaxf(rm, __shfl_xor(rm, 4, 32));
      rm = fmaxf(rm, __shfl_xor(rm, 8, 32));
      const float mn = fmaxf(m[r], rm);
      const float pv = __expf(sv - mn);
      float rs = pv;                               // row sum over 16-lane half
      rs += __shfl_xor(rs, 1, 32);
      rs += __shfl_xor(rs, 2, 32);
      rs += __shfl_xor(rs, 4, 32);
      rs += __shfl_xor(rs, 8, 32);
      const float al = __expf(m[r] - mn);
      l[r] = l[r] * al + rs;
      m[r] = mn;
      alpha[r] = al;
      lp[wave][(mb + r) * 16 + n] = pv;            // stash P for layout swap
    }
    asm volatile("s_wait_dscnt 0" ::: "memory");   // P visible across lanes

#pragma unroll
    for (int r = 0; r < 8; ++r) {                  // rescale O by alpha
      o0[r] *= alpha[r]; o1[r] *= alpha[r]; o2[r] *= alpha[r]; o3[r] *= alpha[r];
    }

    // ---- O += P * V  (16 wmma: 4 K-chunks x 4 col tiles) ----
#pragma unroll
    for (int k0 = 0; k0 < 16; k0 += 4) {
      v2f a  = frag_nk(lp[wave], 16, k0);
      v2f b0 = frag_kn(lvj, H, k0, 0);
      v2f b1 = frag_kn(lvj, H, k0, 16);
      v2f b2 = frag_kn(lvj, H, k0, 32);
      v2f b3 = frag_kn(lvj, H, k0, 48);
      o0 = __builtin_amdgcn_wmma_f32_16x16x4_f32(false, a, false, b0, (short)0, o0, false, false);
      o1 = __builtin_amdgcn_wmma_f32_16x16x4_f32(false, a, false, b1, (short)0, o1, false, false);
      o2 = __builtin_amdgcn_wmma_f32_16x16x4_f32(false, a, false, b2, (short)0, o2, false, false);
      o3 = __builtin_amdgcn_wmma_f32_16x16x4_f32(false, a, false, b3, (short)0, o3, false, false);
    }
  }

  // ---- epilogue: O / l ----
  const int n  = lane & 15;
  const int mb = (lane >> 4) << 3;
  float* __restrict__ ob = out + (size_t)b * T * H;
#pragma unroll
  for (int r = 0; r < 8; ++r) {
    const float inv = 1.0f / l[r];
    const size_t row = (size_t)(qrow0 + mb + r) * H;
    ob[row + 0  + n] = o0[r] * inv;
    ob[row + 16 + n] = o1[r] * inv;
    ob[row + 32 + n] = o2[r] * inv;
    ob[row + 48 + n] = o3[r] * inv;
  }
}

// ---------------------------------------------------------------------------
extern "C" void kernel_launch(void* const* d_in, const int* in_sizes, int n_in,
                              void* d_out, int out_size, void* d_ws, size_t ws_size,
                              hipStream_t stream) {
  (void)in_sizes; (void)n_in; (void)out_size; (void)ws_size;
  const float* x  = (const float*)d_in[0];
  const float* Wq = (const float*)d_in[1];
  const float* Wk = (const float*)d_in[2];
  const float* Wv = (const float*)d_in[3];
  float* out = (float*)d_out;

  float* q  = (float*)d_ws;                       // 3 x 4 MB
  float* k  = q + (size_t)ROWS * H;
  float* v  = k + (size_t)ROWS * H;
  float* Wt = v + (size_t)ROWS * H;               // 3 x 256 KB transposed weights

  wt_kernel<<<(3 * C * H + 255) / 256, 256, 0, stream>>>(Wq, Wk, Wv, Wt);

  qkv_kernel<<<ROWS / 16, 384, 0, stream>>>(x, Wt, q, k, v);

  const int np2 = 2 * ROWS * (H / 2);
  rope_kernel<<<(np2 + 255) / 256, 256, 0, stream>>>(q, k);

  attn_kernel<<<B * (T / (16 * WPB)), 32 * WPB, 0, stream>>>(q, k, v, out);
}